// SoftGrid_79559974191271
// MI455X (gfx1250) — compile-verified
//
#include <hip/hip_runtime.h>
#include <hip/hip_bf16.h>

#define Bsz 64
#define Vn  2000
#define Fn  8
#define Gn  80
#define GF  640          // G*F
#define NRr 5
#define TWO_PI_F 6.283185307179586f
#define EPSf 1e-5f

typedef __attribute__((ext_vector_type(2))) float v2f;
typedef __attribute__((ext_vector_type(8))) float v8f;

// ---------------------------------------------------------------------------
// Kernel A: per (batch, rotation) compute
//   desc[k][b][f*G+g] = (sum_v w[v,g] * feat[b,v,f]) / (sum_v w[v,g] + EPS)
// with w = exp(-(rho-mu_r)^2/(sr^2+eps) - (thmod-mu_t)^2/(st^2+eps)) * mask.
// Block: (80, 3) -> thread.x = gaussian g, thread.y = vertex slice.
// ---------------------------------------------------------------------------
__global__ void __launch_bounds__(240)
softgrid_gauss_desc(const float* __restrict__ feat,
                    const float* __restrict__ rho,
                    const float* __restrict__ theta,
                    const float* __restrict__ mask,
                    const float* __restrict__ mu_rho,
                    const float* __restrict__ sigma_rho,
                    const float* __restrict__ mu_theta,
                    const float* __restrict__ sigma_theta,
                    float* __restrict__ desc)
{
    const int b = blockIdx.x;
    const int k = blockIdx.y;
    const int g = threadIdx.x;       // 0..79
    const int slice = threadIdx.y;   // 0..2

    const float mr  = mu_rho[g];
    const float srs = sigma_rho[g];
    const float inv_r = 1.0f / (srs * srs + EPSf);
    const float mt  = mu_theta[g];
    const float sts = sigma_theta[g];
    const float inv_t = 1.0f / (sts * sts + EPSf);
    const float rot = (float)k * (TWO_PI_F / (float)NRr);

    const float* rb = rho   + (size_t)b * Vn;
    const float* tb = theta + (size_t)b * Vn;
    const float* mb = mask  + (size_t)b * Vn;
    const float* fb = feat  + (size_t)b * Vn * Fn;

    float s = 0.0f;
    float d0 = 0, d1 = 0, d2 = 0, d3 = 0, d4 = 0, d5 = 0, d6 = 0, d7 = 0;

    for (int v = slice; v < Vn; v += 3) {
        const float dr = rb[v] - mr;
        const float th = fmodf(tb[v] + rot, TWO_PI_F);
        const float dt = th - mt;
        const float w  = __expf(-(dr * dr) * inv_r - (dt * dt) * inv_t) * mb[v];
        s += w;
        const float4* fp = (const float4*)(fb + (size_t)v * Fn);
        const float4 f0 = fp[0];
        const float4 f1 = fp[1];
        d0 += w * f0.x;  d1 += w * f0.y;  d2 += w * f0.z;  d3 += w * f0.w;
        d4 += w * f1.x;  d5 += w * f1.y;  d6 += w * f1.z;  d7 += w * f1.w;
    }

    __shared__ float Ssh[3][Gn];
    __shared__ float Dsh[3][Gn][Fn];
    Ssh[slice][g] = s;
    Dsh[slice][g][0] = d0; Dsh[slice][g][1] = d1;
    Dsh[slice][g][2] = d2; Dsh[slice][g][3] = d3;
    Dsh[slice][g][4] = d4; Dsh[slice][g][5] = d5;
    Dsh[slice][g][6] = d6; Dsh[slice][g][7] = d7;
    __syncthreads();

    if (slice == 0) {
        const float stot = Ssh[0][g] + Ssh[1][g] + Ssh[2][g];
        const float inv  = 1.0f / (stot + EPSf);
        float* drow = desc + ((size_t)k * Bsz + b) * GF;
        #pragma unroll
        for (int f = 0; f < Fn; ++f) {
            drow[f * Gn + g] =
                (Dsh[0][g][f] + Dsh[1][g][f] + Dsh[2][g][f]) * inv;
        }
    }
}

// ---------------------------------------------------------------------------
// Kernel B: out[b][j] = relu( max_k ( desc[k][b][:] @ W[:, j] ) + bias[j] )
// One wave (32 lanes) per 16x16 output tile, V_WMMA_F32_16X16X4_F32 over K=640.
// B-tile of W loaded once per k-step and reused for all 5 rotations.
//
// f32 WMMA 16x16x4 VGPR layouts (ISA 7.12.2):
//   A (16x4): lane l<16  -> {A[l][0], A[l][1]},  lane 16+l -> {A[l][2], A[l][3]}
//   B (4x16): lane l<16  -> {B[0][l], B[1][l]},  lane 16+l -> {B[2][l], B[3][l]}
//   D (16x16): VGPR r: lanes 0-15 -> D[r][l], lanes 16-31 -> D[r+8][l]
// ---------------------------------------------------------------------------
__global__ void __launch_bounds__(32)
softgrid_gemm_max_relu(const float* __restrict__ desc,
                       const float* __restrict__ W,
                       const float* __restrict__ bias,
                       float* __restrict__ out)
{
    const int n0   = blockIdx.x * 16;   // output column tile
    const int m0   = blockIdx.y * 16;   // output row (batch) tile
    const int lane = threadIdx.x;
    const int half = lane >> 4;         // 0: K lanes {0,1}; 1: K lanes {2,3}
    const int l    = lane & 15;

    v8f acc0 = {}, acc1 = {}, acc2 = {}, acc3 = {}, acc4 = {};

    for (int kk = 0; kk < GF; kk += 4) {
        const int krow = kk + 2 * half;

        // B-tile from W: rows krow, krow+1; column n0+l
        v2f bm;
        bm.x = W[(size_t)krow * GF + n0 + l];
        bm.y = W[(size_t)(krow + 1) * GF + n0 + l];

        // A-tiles from desc for the 5 rotations (contiguous pair -> b64 load)
        const float* a0p = desc + ((size_t)0 * Bsz + m0 + l) * GF + krow;
        const float* a1p = desc + ((size_t)1 * Bsz + m0 + l) * GF + krow;
        const float* a2p = desc + ((size_t)2 * Bsz + m0 + l) * GF + krow;
        const float* a3p = desc + ((size_t)3 * Bsz + m0 + l) * GF + krow;
        const float* a4p = desc + ((size_t)4 * Bsz + m0 + l) * GF + krow;
        v2f a0 = *(const v2f*)a0p;
        v2f a1 = *(const v2f*)a1p;
        v2f a2 = *(const v2f*)a2p;
        v2f a3 = *(const v2f*)a3p;
        v2f a4 = *(const v2f*)a4p;

        acc0 = __builtin_amdgcn_wmma_f32_16x16x4_f32(false, a0, false, bm,
                                                     (short)0, acc0, false, false);
        acc1 = __builtin_amdgcn_wmma_f32_16x16x4_f32(false, a1, false, bm,
                                                     (short)0, acc1, false, false);
        acc2 = __builtin_amdgcn_wmma_f32_16x16x4_f32(false, a2, false, bm,
                                                     (short)0, acc2, false, false);
        acc3 = __builtin_amdgcn_wmma_f32_16x16x4_f32(false, a3, false, bm,
                                                     (short)0, acc3, false, false);
        acc4 = __builtin_amdgcn_wmma_f32_16x16x4_f32(false, a4, false, bm,
                                                     (short)0, acc4, false, false);
    }

    const float bj = bias[n0 + l];
    #pragma unroll
    for (int r = 0; r < 8; ++r) {
        float m = acc0[r];
        m = fmaxf(m, acc1[r]);
        m = fmaxf(m, acc2[r]);
        m = fmaxf(m, acc3[r]);
        m = fmaxf(m, acc4[r]);
        m = fmaxf(m + bj, 0.0f);
        const int row = m0 + r + (half ? 8 : 0);
        out[(size_t)row * GF + n0 + l] = m;
    }
}

// ---------------------------------------------------------------------------
// Host launcher
// Inputs (setup_inputs order):
//  0 input_feat [B,V,F] f32    1 rho [B,V] f32      2 theta [B,V] f32
//  3 mask [B,V,1] f32          4 mu_rho [G]         5 sigma_rho [G]
//  6 mu_theta [G]              7 sigma_theta [G]
//  8 W [640,640] f32           9 b [640] f32
// Output: [B, 640] f32
// ---------------------------------------------------------------------------
extern "C" void kernel_launch(void* const* d_in, const int* in_sizes, int n_in,
                              void* d_out, int out_size, void* d_ws, size_t ws_size,
                              hipStream_t stream) {
    (void)in_sizes; (void)n_in; (void)out_size; (void)ws_size;
    const float* feat  = (const float*)d_in[0];
    const float* rho   = (const float*)d_in[1];
    const float* theta = (const float*)d_in[2];
    const float* mask  = (const float*)d_in[3];
    const float* mu_r  = (const float*)d_in[4];
    const float* sg_r  = (const float*)d_in[5];
    const float* mu_t  = (const float*)d_in[6];
    const float* sg_t  = (const float*)d_in[7];
    const float* Wm    = (const float*)d_in[8];
    const float* bias  = (const float*)d_in[9];

    float* desc = (float*)d_ws;   // [NR][B][GF] = 5*64*640 f32 = 819200 B

    dim3 gA(Bsz, NRr);
    dim3 bA(Gn, 3, 1);
    softgrid_gauss_desc<<<gA, bA, 0, stream>>>(feat, rho, theta, mask,
                                               mu_r, sg_r, mu_t, sg_t, desc);

    dim3 gB(GF / 16, Bsz / 16);   // (40, 4)
    softgrid_gemm_max_relu<<<gB, dim3(32, 1, 1), 0, stream>>>(desc, Wm, bias,
                                                              (float*)d_out);
}